// LoRAEmbedding_85598698209850
// MI455X (gfx1250) — compile-verified
//
#include <hip/hip_runtime.h>
#include <hip/hip_bf16.h>

// LoRA embedding: out[t,:] = weight[id[t],:] + SCALING * (lora_B @ lora_A[:, id[t]])
// B=4, S=4096 -> 16384 tokens; D=1024; R=8; all fp32.
// Memory-bound (~128 MB traffic @ 23.3 TB/s ~ 5.5us). Rank-8 GEMM done with
// V_WMMA_F32_16X16X4_F32 (the fp32 WMMA shape) so it's free on the matrix pipe
// and matches reference precision exactly.

typedef __attribute__((ext_vector_type(2))) float v2f;
typedef __attribute__((ext_vector_type(8))) float v8f;

#define VOCAB_SZ 128000
#define EMBED_D  1024
#define RANK     8
#define LORA_SCALE 2.0f   // alpha/r = 16/8

__global__ __launch_bounds__(256)
void lora_embed_wmma_kernel(const int* __restrict__ ids,
                            const float* __restrict__ weight,
                            const float* __restrict__ loraA,   // [R, VOCAB]
                            const float* __restrict__ loraB,   // [D, R]
                            float* __restrict__ out,           // [ntok, D]
                            int ntok)
{
    const int lane        = threadIdx.x & 31;
    const int waveInBlk   = threadIdx.x >> 5;
    const int waveGlobal  = blockIdx.x * (blockDim.x >> 5) + waveInBlk;
    const int tbase       = waveGlobal * 16;          // 16 tokens per wave
    if (tbase >= ntok) return;                        // wave-uniform: EXEC stays all-1s below

    const int m  = lane & 15;                         // token row within tile / column within tile
    const int hi = lane >> 4;                         // 0: lanes 0-15, 1: lanes 16-31
    const int k0 = 2 * hi;                            // K layout: vgpr.x=K(2*hi), vgpr.y=K(2*hi+1)

    // token id for A-matrix row (duplicated across wave halves)
    const int id_m = ids[tbase + m];

    // A tiles (16x4 fp32), scaling folded in. Two rank-halves: K = 0..3 and 4..7.
    v2f a0, a1;
    a0.x = LORA_SCALE * loraA[(size_t)(k0    ) * VOCAB_SZ + id_m];
    a0.y = LORA_SCALE * loraA[(size_t)(k0 + 1) * VOCAB_SZ + id_m];
    a1.x = LORA_SCALE * loraA[(size_t)(4 + k0    ) * VOCAB_SZ + id_m];
    a1.y = LORA_SCALE * loraA[(size_t)(4 + k0 + 1) * VOCAB_SZ + id_m];

    // Row ids for the C-layout gather/store: C vgpr i -> row (hi ? 8+i : i).
    int rowid[8];
    int rowtok[8];
#pragma unroll
    for (int i = 0; i < 8; ++i) {
        const int mi = hi ? (8 + i) : i;
        rowtok[i] = tbase + mi;
        rowid[i]  = ids[tbase + mi];                  // same addr across half-wave -> broadcast
    }

    // Loop over 64 column tiles of 16 dims.
    for (int nb = 0; nb < EMBED_D; nb += 16) {
        const int n = nb + m;

        // B tiles (4x16 fp32): B[k][n] = loraB[n*R + k], mirrored K layout.
        v2f b0, b1;
        b0.x = loraB[n * RANK + k0];
        b0.y = loraB[n * RANK + k0 + 1];
        b1.x = loraB[n * RANK + 4 + k0];
        b1.y = loraB[n * RANK + 4 + k0 + 1];

        v8f c = {};
        // D = A1*B1 + (A0*B0 + 0)
        c = __builtin_amdgcn_wmma_f32_16x16x4_f32(false, a0, false, b0,
                                                  (short)0, c, false, false);
        c = __builtin_amdgcn_wmma_f32_16x16x4_f32(false, a1, false, b1,
                                                  (short)0, c, false, false);

        // Add gathered base-embedding rows and store. For each C vgpr i,
        // lanes 0-15 touch weight[id[i]][nb..nb+15] (64B contiguous) and
        // lanes 16-31 touch weight[id[8+i]][nb..nb+15] -> coalesced.
#pragma unroll
        for (int i = 0; i < 8; ++i) {
            const float w = weight[(size_t)rowid[i] * EMBED_D + n];
            out[(size_t)rowtok[i] * EMBED_D + n] = c[i] + w;
        }
    }
}

extern "C" void kernel_launch(void* const* d_in, const int* in_sizes, int n_in,
                              void* d_out, int out_size, void* d_ws, size_t ws_size,
                              hipStream_t stream) {
    const int*   ids    = (const int*)d_in[0];     // input_ids, flat [4*4096]
    const float* weight = (const float*)d_in[1];   // [VOCAB, D]
    const float* loraA  = (const float*)d_in[2];   // [R, VOCAB]
    const float* loraB  = (const float*)d_in[3];   // [D, R]
    float*       out    = (float*)d_out;           // [ntok, D]

    const int ntok  = in_sizes[0];                 // 16384
    const int waves = (ntok + 15) / 16;            // 16 tokens per wave
    const int wavesPerBlock = 8;                   // 256 threads, wave32
    const int blocks = (waves + wavesPerBlock - 1) / wavesPerBlock;

    lora_embed_wmma_kernel<<<blocks, 256, 0, stream>>>(ids, weight, loraA, loraB, out, ntok);
}